// GCN_34351148433642
// MI455X (gfx1250) — compile-verified
//
#include <hip/hip_runtime.h>

#define N_NODES   50000
#define N_EDGES   800000
#define HIDDEN    128
#define N_GRAPHS  256
#define N_CLASSES 10
#define TOT_EDGES (N_EDGES + N_NODES)   // 850000, self loops appended

typedef __attribute__((ext_vector_type(2))) float v2f;
typedef __attribute__((ext_vector_type(8))) float v8f;

// ---------- helpers ----------

// order-preserving float -> uint key (for atomicMax-based segment max)
__device__ __forceinline__ unsigned fkey(float f) {
  unsigned b = __float_as_uint(f);
  return b ^ ((unsigned)((int)b >> 31) | 0x80000000u);
}
__device__ __forceinline__ float fdec(unsigned k) {
  unsigned b = (k & 0x80000000u) ? (k ^ 0x80000000u) : ~k;
  return __uint_as_float(b);
}

__device__ __forceinline__ float wave_sum(float v) {
#pragma unroll
  for (int m = 16; m >= 1; m >>= 1) v += __shfl_xor(v, m, 32);
  return v;
}

__device__ __forceinline__ float lrelu(float v) {
  return v > 0.f ? v : 0.2f * v;
}

// ---------- dual GEMM: O0 = X @ W0, O1 = X @ W1  (fp32 WMMA 16x16x4) ----------
// block = 512 threads = 16 waves; waves 0..7 -> W0/O0 N-tiles, 8..15 -> W1/O1.
// grid.x = M/16 = 3125. No bounds checks needed (exact tiling), EXEC all ones.
__global__ void k_gemm_dual(const float* __restrict__ X,
                            const float* __restrict__ W0,
                            const float* __restrict__ W1,
                            float* __restrict__ O0,
                            float* __restrict__ O1) {
  const int lane = threadIdx.x & 31;
  const int wave = threadIdx.x >> 5;
  const int mt   = blockIdx.x;
  const int mrow = (mt << 4) + (lane & 15);      // A: lanes 0-15 M=0..15, 16-31 same M
  const int koff = (lane >> 4) << 1;             // A: upper half-lanes hold K=2,3
  const float* __restrict__ W = (wave & 8) ? W1 : W0;
  float* __restrict__ O       = (wave & 8) ? O1 : O0;
  const int col = ((wave & 7) << 4) + (lane & 15);

  const float* __restrict__ xrow = X + (size_t)mrow * HIDDEN;
  v8f acc = {};
#pragma unroll
  for (int k0 = 0; k0 < HIDDEN; k0 += 4) {
    v2f a = *(const v2f*)(xrow + k0 + koff);     // even offset -> 8B aligned
    v2f b;
    b.x = W[(size_t)(k0 + koff) * HIDDEN + col];
    b.y = W[(size_t)(k0 + koff + 1) * HIDDEN + col];
    acc = __builtin_amdgcn_wmma_f32_16x16x4_f32(false, a, false, b,
                                                (short)0, acc, false, false);
  }
  // C/D layout: VGPR v -> M = v + (lane>=16 ? 8 : 0), N = lane&15
  const int mbase = (mt << 4) + ((lane >> 4) << 3);
#pragma unroll
  for (int v = 0; v < 8; ++v)
    O[(size_t)(mbase + v) * HIDDEN + col] = acc[v];
}

// ---------- per-edge attention logit: e = att . lrelu(xl[src] + xr[dst]) ----------
// wave per edge, float4 per lane (coalesced 512B per wave)
__global__ void k_edge_logits(const float* __restrict__ XL,
                              const float* __restrict__ XR,
                              const int* __restrict__ ei,
                              const float* __restrict__ att,
                              float* __restrict__ elog) {
  const int e = blockIdx.x * (blockDim.x >> 5) + (threadIdx.x >> 5);
  if (e >= TOT_EDGES) return;
  const int lane = threadIdx.x & 31;
  int s, d;
  if (e < N_EDGES) { s = ei[e]; d = ei[N_EDGES + e]; }
  else             { s = d = e - N_EDGES; }
  const float4 a = ((const float4*)XL)[(size_t)s * 32 + lane];
  const float4 b = ((const float4*)XR)[(size_t)d * 32 + lane];
  const float4 w = ((const float4*)att)[lane];
  float p = w.x * lrelu(a.x + b.x) + w.y * lrelu(a.y + b.y) +
            w.z * lrelu(a.z + b.z) + w.w * lrelu(a.w + b.w);
  p = wave_sum(p);
  if (lane == 0) elog[e] = p;
}

// ---------- segment max over dst (uint-key atomics) ----------
__global__ void k_seg_max(const float* __restrict__ elog,
                          const int* __restrict__ ei,
                          unsigned* __restrict__ emaxk) {
  const int e = blockIdx.x * blockDim.x + threadIdx.x;
  if (e >= TOT_EDGES) return;
  const int d = (e < N_EDGES) ? ei[N_EDGES + e] : (e - N_EDGES);
  atomicMax(&emaxk[d], fkey(elog[e]));
}

// ---------- exp(e - max[dst]) and segment sum ----------
__global__ void k_exp_sum(const float* __restrict__ elog,
                          const int* __restrict__ ei,
                          const unsigned* __restrict__ emaxk,
                          float* __restrict__ eexp,
                          float* __restrict__ den) {
  const int e = blockIdx.x * blockDim.x + threadIdx.x;
  if (e >= TOT_EDGES) return;
  const int d = (e < N_EDGES) ? ei[N_EDGES + e] : (e - N_EDGES);
  const float v = __expf(elog[e] - fdec(emaxk[d]));
  eexp[e] = v;
  atomicAdd(&den[d], v);
}

// ---------- out[dst] += alpha * xl[src]  (wave per edge) ----------
__global__ void k_aggregate(const float* __restrict__ XL,
                            const int* __restrict__ ei,
                            const float* __restrict__ eexp,
                            const float* __restrict__ den,
                            float* __restrict__ agg) {
  const int e = blockIdx.x * (blockDim.x >> 5) + (threadIdx.x >> 5);
  if (e >= TOT_EDGES) return;
  const int lane = threadIdx.x & 31;
  int s, d;
  if (e < N_EDGES) { s = ei[e]; d = ei[N_EDGES + e]; }
  else             { s = d = e - N_EDGES; }
  const float coef = eexp[e] / (den[d] + 1e-16f);
  const float4 a = ((const float4*)XL)[(size_t)s * 32 + lane];
  float* o = agg + (size_t)d * HIDDEN + lane * 4;
  atomicAdd(o + 0, coef * a.x);
  atomicAdd(o + 1, coef * a.y);
  atomicAdd(o + 2, coef * a.z);
  atomicAdd(o + 3, coef * a.w);
}

// ---------- layer-1 epilogue: h = LN(relu(agg + b1)) * g + b  (wave per node) ----------
__global__ void k_post1(const float* __restrict__ agg,
                        const float* __restrict__ b1,
                        const float* __restrict__ gln,
                        const float* __restrict__ bln,
                        float* __restrict__ H) {
  const int n = blockIdx.x * (blockDim.x >> 5) + (threadIdx.x >> 5);
  if (n >= N_NODES) return;
  const int lane = threadIdx.x & 31;
  float4 v  = ((const float4*)agg)[(size_t)n * 32 + lane];
  float4 bb = ((const float4*)b1)[lane];
  v.x = fmaxf(v.x + bb.x, 0.f);
  v.y = fmaxf(v.y + bb.y, 0.f);
  v.z = fmaxf(v.z + bb.z, 0.f);
  v.w = fmaxf(v.w + bb.w, 0.f);
  const float mean = wave_sum(v.x + v.y + v.z + v.w) * (1.f / 128.f);
  const float dx = v.x - mean, dy = v.y - mean, dz = v.z - mean, dw = v.w - mean;
  const float var = wave_sum(dx * dx + dy * dy + dz * dz + dw * dw) * (1.f / 128.f);
  const float inv = rsqrtf(var + 1e-5f);
  const float4 g  = ((const float4*)gln)[lane];
  const float4 bl = ((const float4*)bln)[lane];
  float4 o;
  o.x = dx * inv * g.x + bl.x;
  o.y = dy * inv * g.y + bl.y;
  o.z = dz * inv * g.z + bl.z;
  o.w = dw * inv * g.w + bl.w;
  ((float4*)H)[(size_t)n * 32 + lane] = o;
}

// ---------- layer-2 epilogue + per-graph pooling (wave per node) ----------
__global__ void k_pool(const float* __restrict__ agg,
                       const float* __restrict__ b2,
                       const int* __restrict__ batch,
                       float* __restrict__ pool,
                       float* __restrict__ cnt) {
  const int n = blockIdx.x * (blockDim.x >> 5) + (threadIdx.x >> 5);
  if (n >= N_NODES) return;
  const int lane = threadIdx.x & 31;
  const int g = batch[n];
  const float4 v  = ((const float4*)agg)[(size_t)n * 32 + lane];
  const float4 bb = ((const float4*)b2)[lane];
  float* p = pool + (size_t)g * HIDDEN + lane * 4;
  atomicAdd(p + 0, v.x + bb.x);
  atomicAdd(p + 1, v.y + bb.y);
  atomicAdd(p + 2, v.z + bb.z);
  atomicAdd(p + 3, v.w + bb.w);
  if (lane == 0) atomicAdd(&cnt[g], 1.f);
}

// ---------- classifier: out = (pool/cnt) @ Wlin + blin ----------
__global__ void k_final(const float* __restrict__ pool,
                        const float* __restrict__ cnt,
                        const float* __restrict__ Wlin,
                        const float* __restrict__ blin,
                        float* __restrict__ out) {
  const int g = blockIdx.x;
  __shared__ float sh[HIDDEN];
  const float inv = 1.f / fmaxf(cnt[g], 1.f);
  sh[threadIdx.x] = pool[(size_t)g * HIDDEN + threadIdx.x] * inv;
  __syncthreads();
  if (threadIdx.x < N_CLASSES) {
    float acc = blin[threadIdx.x];
    for (int d = 0; d < HIDDEN; ++d)
      acc += sh[d] * Wlin[d * N_CLASSES + threadIdx.x];
    out[g * N_CLASSES + threadIdx.x] = acc;
  }
}

// ---------- host ----------
extern "C" void kernel_launch(void* const* d_in, const int* in_sizes, int n_in,
                              void* d_out, int out_size, void* d_ws, size_t ws_size,
                              hipStream_t stream) {
  const float* x     = (const float*)d_in[0];
  const int*   ei    = (const int*)  d_in[1];
  const int*   batch = (const int*)  d_in[2];
  const float* Wl1   = (const float*)d_in[3];
  const float* Wr1   = (const float*)d_in[4];
  const float* att1  = (const float*)d_in[5];
  const float* b1    = (const float*)d_in[6];
  const float* gln   = (const float*)d_in[7];
  const float* bln   = (const float*)d_in[8];
  const float* Wl2   = (const float*)d_in[9];
  const float* Wr2   = (const float*)d_in[10];
  const float* att2  = (const float*)d_in[11];
  const float* b2    = (const float*)d_in[12];
  const float* Wlin  = (const float*)d_in[13];
  const float* blin  = (const float*)d_in[14];
  float* out = (float*)d_out;

  char* ws = (char*)d_ws;
  size_t off = 0;
  auto carve = [&](size_t bytes) -> void* {
    void* p = ws + off;
    off += (bytes + 255) & ~(size_t)255;
    return p;
  };
  const size_t nodef = (size_t)N_NODES * HIDDEN * sizeof(float);
  float*    XL    = (float*)   carve(nodef);
  float*    XR    = (float*)   carve(nodef);
  float*    H     = (float*)   carve(nodef);
  float*    AGG   = (float*)   carve(nodef);
  float*    ELOG  = (float*)   carve((size_t)TOT_EDGES * sizeof(float));
  float*    EEXP  = (float*)   carve((size_t)TOT_EDGES * sizeof(float));
  unsigned* EMAXK = (unsigned*)carve((size_t)N_NODES * sizeof(unsigned));
  float*    DEN   = (float*)   carve((size_t)N_NODES * sizeof(float));
  float*    POOL  = (float*)   carve((size_t)N_GRAPHS * HIDDEN * sizeof(float));
  float*    CNT   = (float*)   carve((size_t)N_GRAPHS * sizeof(float));

  const int gM       = N_NODES / 16;                 // 3125, exact
  const int gEdgeW   = TOT_EDGES / 8;                // 106250, exact (wave-per-edge, 256 thr)
  const int gEdgeT   = (TOT_EDGES + 255) / 256;      // thread-per-edge
  const int gNodeW   = N_NODES / 8;                  // 6250, exact

  // ---- layer 1 ----
  k_gemm_dual<<<gM, 512, 0, stream>>>(x, Wl1, Wr1, XL, XR);
  hipMemsetAsync(EMAXK, 0, (size_t)N_NODES * sizeof(unsigned), stream);
  hipMemsetAsync(DEN,   0, (size_t)N_NODES * sizeof(float),    stream);
  hipMemsetAsync(AGG,   0, nodef,                              stream);
  k_edge_logits<<<gEdgeW, 256, 0, stream>>>(XL, XR, ei, att1, ELOG);
  k_seg_max    <<<gEdgeT, 256, 0, stream>>>(ELOG, ei, EMAXK);
  k_exp_sum    <<<gEdgeT, 256, 0, stream>>>(ELOG, ei, EMAXK, EEXP, DEN);
  k_aggregate  <<<gEdgeW, 256, 0, stream>>>(XL, ei, EEXP, DEN, AGG);
  k_post1      <<<gNodeW, 256, 0, stream>>>(AGG, b1, gln, bln, H);

  // ---- layer 2 ----
  k_gemm_dual<<<gM, 512, 0, stream>>>(H, Wl2, Wr2, XL, XR);
  hipMemsetAsync(EMAXK, 0, (size_t)N_NODES * sizeof(unsigned), stream);
  hipMemsetAsync(DEN,   0, (size_t)N_NODES * sizeof(float),    stream);
  hipMemsetAsync(AGG,   0, nodef,                              stream);
  hipMemsetAsync(POOL,  0, (size_t)N_GRAPHS * HIDDEN * sizeof(float), stream);
  hipMemsetAsync(CNT,   0, (size_t)N_GRAPHS * sizeof(float),   stream);
  k_edge_logits<<<gEdgeW, 256, 0, stream>>>(XL, XR, ei, att2, ELOG);
  k_seg_max    <<<gEdgeT, 256, 0, stream>>>(ELOG, ei, EMAXK);
  k_exp_sum    <<<gEdgeT, 256, 0, stream>>>(ELOG, ei, EMAXK, EEXP, DEN);
  k_aggregate  <<<gEdgeW, 256, 0, stream>>>(XL, ei, EEXP, DEN, AGG);

  // ---- pooling + classifier ----
  k_pool <<<gNodeW, 256, 0, stream>>>(AGG, b2, batch, POOL, CNT);
  k_final<<<N_GRAPHS, HIDDEN, 0, stream>>>(POOL, CNT, Wlin, blin, out);
}